// ObjectCondenstationLoss_30021821399476
// MI455X (gfx1250) — compile-verified
//
#include <hip/hip_runtime.h>
#include <math.h>

#define QMIN  0.1f
#define EPSF  1e-7f

// ---------------------------------------------------------------------------
// K1: per-node q = atanh(clip(beta))^2 + qmin ; zero accumulators ; zero keys
// ---------------------------------------------------------------------------
__global__ void k_init(const float* __restrict__ beta,
                       float* __restrict__ q,
                       float* __restrict__ accum,
                       unsigned long long* __restrict__ key,
                       int NN, int NP) {
  int i = blockIdx.x * blockDim.x + threadIdx.x;
  if (i < NN) {
    float cb = fminf(fmaxf(beta[i], EPSF), 1.0f - EPSF);
    float a  = atanhf(cb);
    q[i]     = a * a + QMIN;
    accum[i] = 0.0f;
  }
  if (i < NP) key[i] = 0ull;
}

// ---------------------------------------------------------------------------
// K2: per-particle argmax via packed u64 atomicMax.
//     key = (bits(beta_mail) << 32) | edge_index  (beta_mail >= 0 -> monotone)
//     Ties on beta -> larger edge index wins == reference tie-break.
//     label is true for ~1/NP edges, so atomics + gathers are extremely rare.
// ---------------------------------------------------------------------------
__device__ __forceinline__ void edge_max_one(
    int e, int en, int ep,
    const float* __restrict__ beta,
    const int*   __restrict__ isTrack,
    const int*   __restrict__ parent_target,
    const int*   __restrict__ particle_idx,
    const int*   __restrict__ particle_class,
    unsigned long long* __restrict__ key) {
  int pidx = particle_idx[ep];
  if (parent_target[en] == pidx) {                 // label == 1 (rare)
    float b = beta[en];
    if (particle_class[ep] >= 2) b *= (float)isTrack[en];
    if (b > 0.0f) {
      unsigned long long k =
          ((unsigned long long)__float_as_uint(b) << 32) | (unsigned int)e;
      atomicMax(&key[ep], k);
    }
  }
}

__global__ void k_edge_max(const int4* __restrict__ en4,
                           const int4* __restrict__ ep4,
                           int NE4, int NE,
                           const float* __restrict__ beta,
                           const int*   __restrict__ isTrack,
                           const int*   __restrict__ parent_target,
                           const int*   __restrict__ particle_idx,
                           const int*   __restrict__ particle_class,
                           unsigned long long* __restrict__ key) {
  int stride = gridDim.x * blockDim.x;
  int gid    = blockIdx.x * blockDim.x + threadIdx.x;
  for (int t = gid; t < NE4; t += stride) {
    int4 en = en4[t];
    int4 ep = ep4[t];
    int  e0 = t * 4;
    edge_max_one(e0 + 0, en.x, ep.x, beta, isTrack, parent_target, particle_idx, particle_class, key);
    edge_max_one(e0 + 1, en.y, ep.y, beta, isTrack, parent_target, particle_idx, particle_class, key);
    edge_max_one(e0 + 2, en.z, ep.z, beta, isTrack, parent_target, particle_idx, particle_class, key);
    edge_max_one(e0 + 3, en.w, ep.w, beta, isTrack, parent_target, particle_idx, particle_class, key);
  }
  int rem_start = NE4 * 4;
  int rem = NE - rem_start;
  if (gid < rem) {
    const int* en1 = (const int*)en4;
    const int* ep1 = (const int*)ep4;
    int e = rem_start + gid;
    edge_max_one(e, en1[e], ep1[e], beta, isTrack, parent_target, particle_idx, particle_class, key);
  }
}

// ---------------------------------------------------------------------------
// K3: extract winner per particle -> pq[p] = {max_x.xyz, max_q}
// ---------------------------------------------------------------------------
__global__ void k_extract(const unsigned long long* __restrict__ key,
                          const int*   __restrict__ edge_node,
                          const float* __restrict__ x,
                          const float* __restrict__ q,
                          float4* __restrict__ pq, int NP) {
  int p = blockIdx.x * blockDim.x + threadIdx.x;
  if (p >= NP) return;
  unsigned long long k = key[p];
  float4 r = make_float4(0.0f, 0.0f, 0.0f, 0.0f);
  if ((k >> 32) != 0ull) {                         // max_beta > 0  -> valid
    int e = (int)(k & 0xffffffffull);
    int n = edge_node[e];
    r.x = x[3 * n + 0];
    r.y = x[3 * n + 1];
    r.z = x[3 * n + 2];
    r.w = q[n];
  }
  pq[p] = r;
}

// ---------------------------------------------------------------------------
// K4: potentials. Stage {x,q} table (80KB) + particle_idx (20KB) into LDS
//     with gfx1250 async global->LDS loads, then stream edges as int4.
// ---------------------------------------------------------------------------
typedef int v4i_t __attribute__((vector_size(16)));
#define AS_GLOBAL __attribute__((address_space(1)))
#define AS_LOCAL  __attribute__((address_space(3)))

__global__ void k_edge_pot(const int4* __restrict__ en4,
                           const int4* __restrict__ ep4,
                           int NE4, int NE,
                           const float*  __restrict__ x,
                           const int*    __restrict__ parent_target,
                           const int*    __restrict__ particle_idx,
                           const float4* __restrict__ pq,
                           float* __restrict__ accum, int NP) {
  extern __shared__ char smem[];
  float4* s_pq   = (float4*)smem;
  int*    s_pidx = (int*)(smem + (size_t)NP * sizeof(float4));

#if __has_builtin(__builtin_amdgcn_global_load_async_to_lds_b128) && \
    __has_builtin(__builtin_amdgcn_global_load_async_to_lds_b32)
  for (int i = threadIdx.x; i < NP; i += blockDim.x) {
    __builtin_amdgcn_global_load_async_to_lds_b128(
        (AS_GLOBAL v4i_t*)&pq[i], (AS_LOCAL v4i_t*)&s_pq[i], 0, 0);
    __builtin_amdgcn_global_load_async_to_lds_b32(
        (AS_GLOBAL int*)&particle_idx[i], (AS_LOCAL int*)&s_pidx[i], 0, 0);
  }
  asm volatile("s_wait_asynccnt 0" ::: "memory");
#else
  for (int i = threadIdx.x; i < NP; i += blockDim.x) {
    s_pq[i]   = pq[i];
    s_pidx[i] = particle_idx[i];
  }
#endif
  __syncthreads();

  int stride = gridDim.x * blockDim.x;
  int gid    = blockIdx.x * blockDim.x + threadIdx.x;

  auto body = [&](int en, int ep) {
    float4 w  = s_pq[ep];
    float  mq = w.w;                               // q >= 0.1 when valid; 0 otherwise
    if (mq > 0.0f) {
      bool label = (parent_target[en] == s_pidx[ep]);
      const float* xv = x + 3 * en;
      float dx = w.x - xv[0];
      float dy = w.y - xv[1];
      float dz = w.z - xv[2];
      float d2 = dx * dx + dy * dy + dz * dz;
      float v;
      if (label) {
        v = 3.0f * mq * d2;                        // 3*label*mq*dist^2
      } else {
        float dist = sqrtf(d2);
        v = fmaxf(2.0f - dist, 0.0f) * mq;         // (1-label)*relu(2-dist)*mq
      }
      if (v != 0.0f) atomicAdd(&accum[en], v);
    }
  };

  for (int t = gid; t < NE4; t += stride) {
    int4 en = en4[t];
    int4 ep = ep4[t];
    body(en.x, ep.x);
    body(en.y, ep.y);
    body(en.z, ep.z);
    body(en.w, ep.w);
  }
  int rem_start = NE4 * 4;
  int rem = NE - rem_start;
  if (gid < rem) {
    const int* en1 = (const int*)en4;
    const int* ep1 = (const int*)ep4;
    body(en1[rem_start + gid], ep1[rem_start + gid]);
  }
}

// ---------------------------------------------------------------------------
// K5: out = q * (attract + repulse)
// ---------------------------------------------------------------------------
__global__ void k_final(const float* __restrict__ q,
                        const float* __restrict__ accum,
                        float* __restrict__ out, int NN) {
  int i = blockIdx.x * blockDim.x + threadIdx.x;
  if (i < NN) out[i] = q[i] * accum[i];
}

// ---------------------------------------------------------------------------
extern "C" void kernel_launch(void* const* d_in, const int* in_sizes, int n_in,
                              void* d_out, int out_size, void* d_ws, size_t ws_size,
                              hipStream_t stream) {
  const float* beta           = (const float*)d_in[0];
  const float* x              = (const float*)d_in[1];
  const int*   isTrack        = (const int*)d_in[2];
  const int*   parent_target  = (const int*)d_in[3];
  const int*   particle_idx   = (const int*)d_in[4];
  const int*   particle_class = (const int*)d_in[5];
  const int*   edge_node      = (const int*)d_in[6];
  const int*   edge_particle  = (const int*)d_in[7];
  float*       out            = (float*)d_out;

  const int NN = in_sizes[0];
  const int NP = in_sizes[4];
  const int NE = in_sizes[6];
  const int NE4 = NE / 4;

  // Workspace carve-out (16B aligned sections)
  char* ws = (char*)d_ws;
  size_t off = 0;
  auto carve = [&](size_t bytes) {
    char* p = ws + off;
    off += (bytes + 15) & ~(size_t)15;
    return p;
  };
  float*              q     = (float*)carve((size_t)NN * sizeof(float));
  float*              accum = (float*)carve((size_t)NN * sizeof(float));
  unsigned long long* key   = (unsigned long long*)carve((size_t)NP * sizeof(unsigned long long));
  float4*             pq    = (float4*)carve((size_t)NP * sizeof(float4));
  (void)ws_size;

  const int T = 256;
  const int initBlocks = ( (NN > NP ? NN : NP) + T - 1) / T;

  k_init<<<initBlocks, T, 0, stream>>>(beta, q, accum, key, NN, NP);

  k_edge_max<<<2048, T, 0, stream>>>((const int4*)edge_node, (const int4*)edge_particle,
                                     NE4, NE, beta, isTrack, parent_target,
                                     particle_idx, particle_class, key);

  k_extract<<<(NP + T - 1) / T, T, 0, stream>>>(key, edge_node, x, q, pq, NP);

  size_t smem = (size_t)NP * sizeof(float4) + (size_t)NP * sizeof(int);
  k_edge_pot<<<512, T, smem, stream>>>((const int4*)edge_node, (const int4*)edge_particle,
                                       NE4, NE, x, parent_target, particle_idx,
                                       pq, accum, NP);

  k_final<<<(NN + T - 1) / T, T, 0, stream>>>(q, accum, out, NN);
}